// GNNSat_V2_18940805776101
// MI455X (gfx1250) — compile-verified
//
#include <hip/hip_runtime.h>

#define HID 64
#define EPS 1e-5f

typedef float v2f __attribute__((ext_vector_type(2)));
typedef float v8f __attribute__((ext_vector_type(8)));

__device__ __forceinline__ float leakyf(float x, float s) { return x >= 0.0f ? x : s * x; }

// float atomic max via signed/unsigned int ordering trick (lowers to native
// global_atomic_max / global_atomic_min which the L2 executes at full rate)
__device__ __forceinline__ void atomicMaxF(float* addr, float v) {
    if (v >= 0.0f) atomicMax((int*)addr, __float_as_int(v));
    else           atomicMin((unsigned int*)addr, __float_as_uint(v));
}

__device__ __forceinline__ float wredsum(float v) {
#pragma unroll
    for (int o = 16; o > 0; o >>= 1) v += __shfl_xor(v, o, 32);
    return v;
}

// ---------------- utility ----------------
__global__ void gat_fill(float* p, float v, int n) {
    int i = blockIdx.x * blockDim.x + threadIdx.x;
    if (i < n) p[i] = v;
}

// degree count + edge_attr sums per destination (self-loop fill_value='mean')
__global__ void gat_deg(const int* __restrict__ src, const int* __restrict__ dst,
                        const float* __restrict__ ea, float* cnt, float* sumattr, int E) {
    int e = blockIdx.x * blockDim.x + threadIdx.x;
    if (e >= E) return;
    int d = dst[e];
    (void)src;
    atomicAdd(&cnt[d], 1.0f);
    atomicAdd(&sumattr[2 * d + 0], ea[2 * e + 0]);
    atomicAdd(&sumattr[2 * d + 1], ea[2 * e + 1]);
}

__global__ void gat_loopattr(float* sumattr, const float* __restrict__ cnt, int n) {
    int i = blockIdx.x * blockDim.x + threadIdx.x;
    if (i >= n) return;
    float c = fmaxf(cnt[i], 1.0f);
    sumattr[2 * i + 0] /= c;
    sumattr[2 * i + 1] /= c;
}

// ce[0..1] = We1 @ att_e1 ; ce[2..3] = We2 @ att_e2  (collapse of edge MLP)
__global__ void gat_ce(const float* __restrict__ We1, const float* __restrict__ ate1,
                       const float* __restrict__ We2, const float* __restrict__ ate2,
                       float* ce) {
    int i = threadIdx.x;
    if (i >= 4) return;
    const float* Wr = (i < 2) ? (We1 + i * HID) : (We2 + (i - 2) * HID);
    const float* av = (i < 2) ? ate1 : ate2;
    float acc = 0.0f;
    for (int j = 0; j < HID; ++j) acc += Wr[j] * av[j];
    ce[i] = acc;
}

// layer-1 transform: h = x @ W1 (Fin=2) + attention dots, one wave per node
__global__ void gat_h1(const float* __restrict__ x, const float* __restrict__ W1,
                       const float* __restrict__ atts, const float* __restrict__ attd,
                       float* __restrict__ h, float* __restrict__ as_, float* __restrict__ ad_,
                       int n) {
    int node = blockIdx.x * 8 + (threadIdx.x >> 5);
    if (node >= n) return;
    int lane = threadIdx.x & 31;
    float x0 = x[2 * node], x1 = x[2 * node + 1];
    float h0 = x0 * W1[lane] + x1 * W1[HID + lane];
    float h1 = x0 * W1[lane + 32] + x1 * W1[HID + lane + 32];
    h[(size_t)node * HID + lane] = h0;
    h[(size_t)node * HID + lane + 32] = h1;
    float ps = wredsum(h0 * atts[lane] + h1 * atts[lane + 32]);
    float pd = wredsum(h0 * attd[lane] + h1 * attd[lane + 32]);
    if (lane == 0) { as_[node] = ps; ad_[node] = pd; }
}

// edge pass 1: logits (a_e computed on the fly) + segment max (float atomic max)
__global__ void gat_edge_max(const int* __restrict__ src, const int* __restrict__ dst,
                             const float* __restrict__ as_, const float* __restrict__ ad_,
                             const float* __restrict__ ea, const float* __restrict__ loopattr,
                             const float* __restrict__ ce, float* __restrict__ lg,
                             float* __restrict__ m, int E, int Etot) {
    int e = blockIdx.x * blockDim.x + threadIdx.x;
    if (e >= Etot) return;
    int s_, d_;
    float f0, f1;
    if (e < E) { s_ = src[e]; d_ = dst[e]; f0 = ea[2 * e]; f1 = ea[2 * e + 1]; }
    else { s_ = d_ = e - E; f0 = loopattr[2 * s_]; f1 = loopattr[2 * s_ + 1]; }
    float l = as_[s_] + ad_[d_] + f0 * ce[0] + f1 * ce[1];
    l = leakyf(l, 0.2f);
    lg[e] = l;
    atomicMaxF(&m[d_], l);
}

// edge pass 2: p = exp(logit - m[dst]); store p over logits; segment sum
__global__ void gat_edge_expsum(const int* __restrict__ dst, float* __restrict__ lg,
                                const float* __restrict__ m, float* __restrict__ s,
                                int E, int Etot) {
    int e = blockIdx.x * blockDim.x + threadIdx.x;
    if (e >= Etot) return;
    int d_ = (e < E) ? dst[e] : (e - E);
    float p = __expf(lg[e] - m[d_]);
    lg[e] = p;
    atomicAdd(&s[d_], p);
}

// edge pass 3: out[dst] += h[src] * (p / s[dst]); 16 threads/edge, float4 loads
__global__ void gat_edge_aggr(const int* __restrict__ src, const int* __restrict__ dst,
                              const float* __restrict__ h, const float* __restrict__ p,
                              const float* __restrict__ s, float* __restrict__ out,
                              int E, int Etot) {
    long long t = (long long)blockIdx.x * blockDim.x + threadIdx.x;
    int e = (int)(t >> 4);
    if (e >= Etot) return;
    int q = (int)(t & 15);
    int s_, d_;
    if (e < E) { s_ = src[e]; d_ = dst[e]; } else { s_ = d_ = e - E; }
    float alpha = p[e] / s[d_];
    const float4 hv = *(const float4*)(h + (size_t)s_ * HID + q * 4);
    float* o = out + (size_t)d_ * HID + q * 4;
    atomicAdd(o + 0, hv.x * alpha);
    atomicAdd(o + 1, hv.y * alpha);
    atomicAdd(o + 2, hv.z * alpha);
    atomicAdd(o + 3, hv.w * alpha);
}

// batchnorm stats: per-channel sum / sumsq (bias folded in)
__global__ void gat_bnstats(const float* __restrict__ a, const float* __restrict__ b1,
                            float* sum, float* sumsq, int n) {
    int tid = blockIdx.x * blockDim.x + threadIdx.x;
    int c = tid & 63;
    int stride = (gridDim.x * blockDim.x) >> 6;
    float bc = b1[c], ls = 0.0f, lq = 0.0f;
    for (int r = tid >> 6; r < n; r += stride) {
        float v = a[(size_t)r * HID + c] + bc;
        ls += v; lq += v * v;
    }
    atomicAdd(&sum[c], ls);
    atomicAdd(&sumsq[c], lq);
}

__global__ void gat_bnfin(const float* sum, const float* sumsq, float* mu, float* rstd, int n) {
    int c = threadIdx.x;
    if (c >= HID) return;
    float m = sum[c] / (float)n;
    float var = sumsq[c] / (float)n - m * m;
    mu[c] = m;
    rstd[c] = rsqrtf(var + EPS);
}

__global__ void gat_bnapply(const float* __restrict__ a, const float* __restrict__ b1,
                            const float* __restrict__ mu, const float* __restrict__ rstd,
                            const float* __restrict__ gm, const float* __restrict__ bt,
                            float* __restrict__ o, int total) {
    int i = blockIdx.x * blockDim.x + threadIdx.x;
    if (i >= total) return;
    int c = i & 63;
    float v = (a[i] + b1[c] - mu[c]) * rstd[c] * gm[c] + bt[c];
    o[i] = leakyf(v, 0.01f);
}

// -------- layer-2 dense transform: h2 = hin[N,64] @ W2[64,64] via f32 WMMA --------
// Per ISA 7.12.2: A 16x4 f32: lanes0-15 hold {K0,K1}, lanes16-31 {K2,K3};
// C/D 16x16 f32: VGPR v = row v (lanes0-15) / row v+8 (lanes16-31), N=lane&15.
__global__ void gat_h2_wmma(const float* __restrict__ hin, const float* __restrict__ W2,
                            float* __restrict__ hout) {
    int wave = threadIdx.x >> 5;                   // 8 waves: 2 node-tiles x 4 col-tiles
    int nodeTile = blockIdx.x * 32 + (wave >> 2) * 16;
    int colBase = (wave & 3) * 16;
    int lane = threadIdx.x & 31;
    int l15 = lane & 15;
    int half = lane >> 4;
    const float* arow = hin + (size_t)(nodeTile + l15) * HID;
#if __has_builtin(__builtin_amdgcn_wmma_f32_16x16x4_f32)
    v8f acc = {};
#pragma unroll
    for (int kk = 0; kk < 16; ++kk) {
        int k0 = kk * 4 + 2 * half;
        v2f a; a.x = arow[k0]; a.y = arow[k0 + 1];
        v2f b; b.x = W2[k0 * HID + colBase + l15]; b.y = W2[(k0 + 1) * HID + colBase + l15];
        acc = __builtin_amdgcn_wmma_f32_16x16x4_f32(false, a, false, b, (short)0, acc,
                                                    false, false);
    }
#pragma unroll
    for (int v = 0; v < 8; ++v)
        hout[(size_t)(nodeTile + v + 8 * half) * HID + colBase + l15] = acc[v];
#else
    // scalar fallback (keeps compile green if the f32 WMMA builtin is absent)
    for (int v = 0; v < 8; ++v) {
        int row = nodeTile + v + 8 * half;
        int col = colBase + l15;
        float acc = 0.0f;
        for (int k = 0; k < HID; ++k) acc = fmaf(hin[(size_t)row * HID + k], W2[k * HID + col], acc);
        hout[(size_t)row * HID + col] = acc;
    }
#endif
}

// scalar tail for N not divisible by 32 (not hit for N=100000)
__global__ void gat_h2_tail(const float* __restrict__ hin, const float* __restrict__ W2,
                            float* __restrict__ hout, int start, int n) {
    int idx = blockIdx.x * blockDim.x + threadIdx.x;
    int node = start + idx / HID;
    if (node >= n) return;
    int col = idx & 63;
    float acc = 0.0f;
    for (int k = 0; k < HID; ++k) acc = fmaf(hin[(size_t)node * HID + k], W2[k * HID + col], acc);
    hout[(size_t)node * HID + col] = acc;
}

// attention dots for layer 2 (wave per node)
__global__ void gat_attn2(const float* __restrict__ h, const float* __restrict__ atts,
                          const float* __restrict__ attd, float* __restrict__ as_,
                          float* __restrict__ ad_, int n) {
    int node = blockIdx.x * 8 + (threadIdx.x >> 5);
    if (node >= n) return;
    int lane = threadIdx.x & 31;
    float h0 = h[(size_t)node * HID + lane];
    float h1 = h[(size_t)node * HID + lane + 32];
    float ps = wredsum(h0 * atts[lane] + h1 * atts[lane + 32]);
    float pd = wredsum(h0 * attd[lane] + h1 * attd[lane + 32]);
    if (lane == 0) { as_[node] = ps; ad_[node] = pd; }
}

__global__ void gat_post2(float* a, const float* __restrict__ b2, int total) {
    int i = blockIdx.x * blockDim.x + threadIdx.x;
    if (i >= total) return;
    a[i] = leakyf(a[i] + b2[i & 63], 0.01f);
}

// final MLP 64->32->1 * mask; one wave/node, lane = hidden unit, Wf1 staged in LDS
__global__ void gat_mlp(const float* __restrict__ h, const float* __restrict__ Wf1,
                        const float* __restrict__ bf1, const float* __restrict__ Wf2,
                        const float* __restrict__ bf2, const float* __restrict__ mask,
                        float* __restrict__ out, int n) {
    __shared__ float sW[HID * 32];
    __shared__ float sb1[32];
    __shared__ float sW2[32];
    for (int i = threadIdx.x; i < HID * 32; i += blockDim.x) sW[i] = Wf1[i];
    if (threadIdx.x < 32) { sb1[threadIdx.x] = bf1[threadIdx.x]; sW2[threadIdx.x] = Wf2[threadIdx.x]; }
    __syncthreads();
    int lane = threadIdx.x & 31;
    for (int node = blockIdx.x * 8 + (threadIdx.x >> 5); node < n; node += gridDim.x * 8) {
        const float* hr = h + (size_t)node * HID;
        float acc = sb1[lane];
#pragma unroll
        for (int j = 0; j < HID; ++j) acc = fmaf(hr[j], sW[j * 32 + lane], acc);
        float z = leakyf(acc, 0.01f) * sW2[lane];
        z = wredsum(z);
        if (lane == 0) out[node] = (z + bf2[0]) * mask[node];
    }
}

static inline int cdiv(long long a, int b) { return (int)((a + b - 1) / b); }

extern "C" void kernel_launch(void* const* d_in, const int* in_sizes, int n_in,
                              void* d_out, int out_size, void* d_ws, size_t ws_size,
                              hipStream_t stream) {
    (void)n_in; (void)out_size; (void)ws_size;
    const float* x    = (const float*)d_in[0];
    const int*   ei   = (const int*)  d_in[1];
    const float* ea   = (const float*)d_in[2];
    const float* mask = (const float*)d_in[3];
    const float* W1   = (const float*)d_in[4];
    const float* as1w = (const float*)d_in[5];
    const float* ad1w = (const float*)d_in[6];
    const float* We1  = (const float*)d_in[7];
    const float* ae1w = (const float*)d_in[8];
    const float* b1   = (const float*)d_in[9];
    const float* W2   = (const float*)d_in[10];
    const float* as2w = (const float*)d_in[11];
    const float* ad2w = (const float*)d_in[12];
    const float* We2  = (const float*)d_in[13];
    const float* ae2w = (const float*)d_in[14];
    const float* b2   = (const float*)d_in[15];
    const float* gm   = (const float*)d_in[16];
    const float* bt   = (const float*)d_in[17];
    const float* Wf1  = (const float*)d_in[18];
    const float* bf1  = (const float*)d_in[19];
    const float* Wf2  = (const float*)d_in[20];
    const float* bf2  = (const float*)d_in[21];

    const int N = in_sizes[0] / 2;
    const int E = in_sizes[1] / 2;
    const int Etot = E + N;
    const int* srcA = ei;
    const int* dstA = ei + E;

    float* W = (float*)d_ws;
    size_t off = 0;
    float* B0 = W + off; off += (size_t)N * HID;   // h1 -> reused as hin2
    float* B1 = W + off; off += (size_t)N * HID;   // aggregation target (both layers)
    float* B2 = W + off; off += (size_t)N * HID;   // h2pre
    float* lg = W + off; off += (size_t)Etot;      // logits, then p in place
    float* a_s1 = W + off; off += N;
    float* a_d1 = W + off; off += N;
    float* a_s2 = W + off; off += N;
    float* a_d2 = W + off; off += N;
    float* mseg = W + off; off += N;               // segment max
    float* sseg = W + off; off += N;               // segment sum
    float* cnt  = W + off; off += N;
    float* lattr= W + off; off += (size_t)2 * N;   // sumattr -> loop_attr in place
    float* bns  = W + off; off += 64;              // channel sums
    float* bnq  = W + off; off += 64;              // channel sumsq
    float* muv  = W + off; off += 64;
    float* rstd = W + off; off += 64;
    float* ce   = W + off; off += 4;

    const int TB = 256;
    float* outp = (float*)d_out;

    // self-loop mean attrs
    gat_fill<<<cdiv(N, TB), TB, 0, stream>>>(cnt, 0.0f, N);
    gat_fill<<<cdiv(2 * N, TB), TB, 0, stream>>>(lattr, 0.0f, 2 * N);
    gat_deg<<<cdiv(E, TB), TB, 0, stream>>>(srcA, dstA, ea, cnt, lattr, E);
    gat_loopattr<<<cdiv(N, TB), TB, 0, stream>>>(lattr, cnt, N);
    gat_ce<<<1, 64, 0, stream>>>(We1, ae1w, We2, ae2w, ce);

    // ---- layer 1 ----
    gat_h1<<<cdiv(N, 8), TB, 0, stream>>>(x, W1, as1w, ad1w, B0, a_s1, a_d1, N);
    gat_fill<<<cdiv(N, TB), TB, 0, stream>>>(mseg, -3.0e38f, N);
    gat_fill<<<cdiv(N, TB), TB, 0, stream>>>(sseg, 0.0f, N);
    gat_fill<<<cdiv((long long)N * HID, TB), TB, 0, stream>>>(B1, 0.0f, N * HID);
    gat_edge_max<<<cdiv(Etot, TB), TB, 0, stream>>>(srcA, dstA, a_s1, a_d1, ea, lattr, ce,
                                                    lg, mseg, E, Etot);
    gat_edge_expsum<<<cdiv(Etot, TB), TB, 0, stream>>>(dstA, lg, mseg, sseg, E, Etot);
    gat_edge_aggr<<<cdiv((long long)Etot * 16, TB), TB, 0, stream>>>(srcA, dstA, B0, lg, sseg,
                                                                     B1, E, Etot);
    // batchnorm + leaky
    gat_fill<<<1, 128, 0, stream>>>(bns, 0.0f, 128);
    gat_bnstats<<<128, TB, 0, stream>>>(B1, b1, bns, bnq, N);
    gat_bnfin<<<1, 64, 0, stream>>>(bns, bnq, muv, rstd, N);
    gat_bnapply<<<cdiv((long long)N * HID, TB), TB, 0, stream>>>(B1, b1, muv, rstd, gm, bt,
                                                                 B0, N * HID);

    // ---- layer 2 ----
    int nt32 = N / 32;
    if (nt32 > 0) gat_h2_wmma<<<nt32, TB, 0, stream>>>(B0, W2, B2);
    int tail = N - nt32 * 32;
    if (tail > 0)
        gat_h2_tail<<<cdiv((long long)tail * HID, TB), TB, 0, stream>>>(B0, W2, B2, nt32 * 32, N);
    gat_attn2<<<cdiv(N, 8), TB, 0, stream>>>(B2, as2w, ad2w, a_s2, a_d2, N);
    gat_fill<<<cdiv(N, TB), TB, 0, stream>>>(mseg, -3.0e38f, N);
    gat_fill<<<cdiv(N, TB), TB, 0, stream>>>(sseg, 0.0f, N);
    gat_fill<<<cdiv((long long)N * HID, TB), TB, 0, stream>>>(B1, 0.0f, N * HID);
    gat_edge_max<<<cdiv(Etot, TB), TB, 0, stream>>>(srcA, dstA, a_s2, a_d2, ea, lattr, ce + 2,
                                                    lg, mseg, E, Etot);
    gat_edge_expsum<<<cdiv(Etot, TB), TB, 0, stream>>>(dstA, lg, mseg, sseg, E, Etot);
    gat_edge_aggr<<<cdiv((long long)Etot * 16, TB), TB, 0, stream>>>(srcA, dstA, B2, lg, sseg,
                                                                     B1, E, Etot);
    gat_post2<<<cdiv((long long)N * HID, TB), TB, 0, stream>>>(B1, b2, N * HID);

    // ---- final MLP ----
    int mlpBlocks = cdiv(N, 8) < 1024 ? cdiv(N, 8) : 1024;
    gat_mlp<<<mlpBlocks, TB, 0, stream>>>(B1, Wf1, bf1, Wf2, bf2, mask, outp, N);
}